// PtychographyModel_8100308320928
// MI455X (gfx1250) — compile-verified
//
#include <hip/hip_runtime.h>
#include <math.h>

// ---------------------------------------------------------------------------
// Ptychography multislice forward model for MI455X (gfx1250, wave32).
// FFT-256 factored radix-16 x radix-16; each 16-point DFT stage is a
// 16x16x16 complex matmul on V_WMMA_F32_16X16X4_F32 (fp32 matrix pipe).
// DFT-matrix fragments + twiddles live in per-block LDS tables with BOTH
// signs precomputed (fwd/inv), so the inner FFT does zero sign VALU work.
// All intermediate images are L2-resident (192MB L2).
// ---------------------------------------------------------------------------

#define NP       256
#define NPP      (NP * NP)
#define NPAD     257          // LDS tile pitch: avoids bank conflicts
#define NOBJ     1200
#define NSLICES  8
#define NMODES   4
#define BATCH    64
#define PIXELSZ  0.05f
#define WAVELENF 0.0251f
#define DZF      2.0f
#define PI_F     3.14159265358979f

typedef float v2f __attribute__((ext_vector_type(2)));
typedef float v8f __attribute__((ext_vector_type(8)));

__device__ __forceinline__ void wave_dswait() {
    // intra-wave LDS ordering between FFT relayout stages
    asm volatile("s_wait_dscnt 0" ::: "memory");
}

__device__ __forceinline__ float sfreq(int i) {
    const float dk = 1.0f / (NP * PIXELSZ);
    int si = (i < NP / 2) ? i : i - NP;
    return dk * (float)si;
}

__device__ __forceinline__ v8f wmma_f32(v2f a, v2f b, v8f c) {
    return __builtin_amdgcn_wmma_f32_16x16x4_f32(
        false, a, false, b, (short)0, c, false, false);
}

// ---------------------------------------------------------------------------
// Per-block constant tables, both transform signs precomputed:
//   fFre[kk][lane]      = F16 fragment cos for rows/cols (k0,k0+1), k0=4kk+2hi
//   fFim[s][kk][lane]   = F16 fragment sin, s=0 fwd (-), s=1 inv (+)
//                         (F16 symmetric -> serves step-1 A-frag AND step-3 B-frag;
//                          opposite-sign table IS the negated fragment)
//   twT[s][v][lane]     = W256 twiddle {cos, sgn*sin}
// ---------------------------------------------------------------------------
__device__ void fft_tables_init(float2 (*fFre)[32], float2 (*fFim)[4][32],
                                float2 (*twT)[8][32])
{
    const int tid = threadIdx.x;
    const float w16  = -2.0f * PI_F / 16.0f;
    const float w256 = -2.0f * PI_F / 256.0f;
    if (tid < 128) {
        const int kk = tid >> 5, ln = tid & 31;
        const int nlo = ln & 15, hi = ln >> 4;
        const int k0 = 4 * kk + 2 * hi;
        float s0, c0, s1, c1;
        __sincosf(w16 * (float)(nlo * k0), &s0, &c0);
        __sincosf(w16 * (float)(nlo * (k0 + 1)), &s1, &c1);
        fFre[kk][ln]    = make_float2(c0, c1);
        fFim[0][kk][ln] = make_float2(s0, s1);      // forward
        fFim[1][kk][ln] = make_float2(-s0, -s1);    // inverse (conjugate)
    }
    if (tid < 256) {
        const int v = tid >> 5, ln = tid & 31;
        const int nlo = ln & 15, hi = ln >> 4;
        float s, c;
        __sincosf(w256 * (float)((v + 8 * hi) * nlo), &s, &c);
        twT[0][v][ln] = make_float2(c, s);
        twT[1][v][ln] = make_float2(c, -s);
    }
    __syncthreads();
}

// ---------------------------------------------------------------------------
// One wave: 256-point complex FFT of its LDS tile (x[n] at index n, n=16n1+n2).
//   step1: D1 = F16 * X       (A = F from tables, B = X from tile)
//   twiddle W256, relayout through tile
//   step3: D2 = D1' * F16     (A = D1' from tile, B = F from tables)
// fImA: sign-selected sin table; fImN: opposite sign (== negated fragment).
// ---------------------------------------------------------------------------
__device__ void fft256_wave(float* __restrict__ tre, float* __restrict__ tim,
                            const float2 (*fFre)[32],
                            const float2 (*fImA)[32],
                            const float2 (*fImN)[32],
                            const float2 (*twS)[32],
                            float scale, int lane)
{
    const int nlo = lane & 15;
    const int hi  = lane >> 4;

    v8f cr = {}; v8f ci = {};
#pragma unroll
    for (int kk = 0; kk < 4; ++kk) {
        const int k0 = 4 * kk + 2 * hi;
        float2 qr = fFre[kk][lane];
        float2 qi = fImA[kk][lane];
        float2 qn = fImN[kk][lane];
        v2f ar; ar.x = qr.x; ar.y = qr.y;
        v2f ai; ai.x = qi.x; ai.y = qi.y;
        v2f an; an.x = qn.x; an.y = qn.y;
        v2f bxr; bxr.x = tre[k0 * 16 + nlo]; bxr.y = tre[(k0 + 1) * 16 + nlo];
        v2f bxi; bxi.x = tim[k0 * 16 + nlo]; bxi.y = tim[(k0 + 1) * 16 + nlo];
        cr = wmma_f32(ar, bxr, cr);
        cr = wmma_f32(an, bxi, cr);
        ci = wmma_f32(ar, bxi, ci);
        ci = wmma_f32(ai, bxr, ci);
    }

    wave_dswait();
#pragma unroll
    for (int v = 0; v < 8; ++v) {
        const int m = v + 8 * hi;
        float2 t = twS[v][lane];              // {c, sgn*s}
        float dr = cr[v] * t.x - ci[v] * t.y;
        float di = cr[v] * t.y + ci[v] * t.x;
        tre[m * 16 + nlo] = dr;
        tim[m * 16 + nlo] = di;
    }
    wave_dswait();

    v8f dr2 = {}; v8f di2 = {};
#pragma unroll
    for (int kk = 0; kk < 4; ++kk) {
        const int k0 = 4 * kk + 2 * hi;
        v2f axr; axr.x = tre[nlo * 16 + k0]; axr.y = tre[nlo * 16 + k0 + 1];
        v2f axi; axi.x = tim[nlo * 16 + k0]; axi.y = tim[nlo * 16 + k0 + 1];
        v2f axn; axn.x = -axi.x;             axn.y = -axi.y;
        float2 qr = fFre[kk][lane];
        float2 qi = fImA[kk][lane];
        v2f bfr; bfr.x = qr.x; bfr.y = qr.y;
        v2f bfi; bfi.x = qi.x; bfi.y = qi.y;
        dr2 = wmma_f32(axr, bfr, dr2);
        dr2 = wmma_f32(axn, bfi, dr2);
        di2 = wmma_f32(axr, bfi, di2);
        di2 = wmma_f32(axi, bfr, di2);
    }

    wave_dswait();
#pragma unroll
    for (int v = 0; v < 8; ++v) {
        const int k1 = v + 8 * hi;
        tre[k1 + 16 * nlo] = dr2[v] * scale;
        tim[k1 + 16 * nlo] = di2[v] * scale;
    }
    wave_dswait();
}

// ---------------------------------------------------------------------------
// In-place 2-D FFT of a 256x256 complex image (interleaved float2, L2
// resident). Rows: per-wave contiguous (coalesced). Columns: 512-thread
// cooperative transposed staging (128B-contiguous global access) into the
// 16 padded per-wave tiles, one column FFT per wave.
// sel: 0 = forward, 1 = inverse (conjugated tables); scale on column pass.
// ---------------------------------------------------------------------------
__device__ void fft2_block(float2* __restrict__ img, int sel, float scale,
                           float (*tileRe)[NPAD], float (*tileIm)[NPAD],
                           const float2 (*fFre)[32],
                           const float2 (*fFim)[4][32],
                           const float2 (*twT)[8][32])
{
    const int tid    = threadIdx.x;
    const int lane   = tid & 31;
    const int wid    = tid >> 5;
    const int nwaves = blockDim.x >> 5;   // 16
    const float2 (*fImA)[32] = fFim[sel];
    const float2 (*fImN)[32] = fFim[sel ^ 1];   // negated fragment
    const float2 (*twS)[32]  = twT[sel];
    float* tre = tileRe[wid];
    float* tim = tileIm[wid];

    // ---- row pass (in place) ----
    for (int r = wid; r < NP; r += nwaves) {
#pragma unroll
        for (int j = 0; j < 8; ++j) {
            float2 v = img[r * NP + lane + 32 * j];
            tre[lane + 32 * j] = v.x;
            tim[lane + 32 * j] = v.y;
        }
        wave_dswait();
        fft256_wave(tre, tim, fFre, fImA, fImN, twS, 1.0f, lane);
#pragma unroll
        for (int j = 0; j < 8; ++j) {
            float2 v;
            v.x = tre[lane + 32 * j];
            v.y = tim[lane + 32 * j];
            img[r * NP + lane + 32 * j] = v;
        }
    }
    __threadfence_block();
    __syncthreads();

    // ---- column pass: cooperative transposed staging, 16 columns/iter ----
    for (int c0 = 0; c0 < NP; c0 += 16) {
#pragma unroll
        for (int it = 0; it < 8; ++it) {
            const int r  = it * 32 + (tid >> 4);
            const int cc = tid & 15;
            float2 v = img[r * NP + c0 + cc];
            tileRe[cc][r] = v.x;
            tileIm[cc][r] = v.y;
        }
        __syncthreads();
        fft256_wave(tileRe[wid], tileIm[wid], fFre, fImA, fImN, twS, scale, lane);
        __syncthreads();
#pragma unroll
        for (int it = 0; it < 8; ++it) {
            const int r  = it * 32 + (tid >> 4);
            const int cc = tid & 15;
            float2 v;
            v.x = tileRe[cc][r];
            v.y = tileIm[cc][r];
            img[r * NP + c0 + cc] = v;
        }
        __syncthreads();
    }
    __threadfence_block();
    __syncthreads();
}

// ---------------------------------------------------------------------------
__global__ void init_kernel(float* __restrict__ out, int n,
                            float2* __restrict__ propT)
{
    for (int i = blockIdx.x * blockDim.x + threadIdx.x; i < n;
         i += gridDim.x * blockDim.x)
        out[i] = 0.0f;
    // Fresnel propagator table (position-independent, L2 resident)
    for (int p = blockIdx.x * blockDim.x + threadIdx.x; p < NPP;
         p += gridDim.x * blockDim.x) {
        const int i = p >> 8, j = p & 255;
        const float fx = sfreq(i), fy = sfreq(j);
        float s, c;
        __sincosf(-PI_F * WAVELENF * (fx * fx + fy * fy) * DZF, &s, &c);
        propT[p] = make_float2(c, s);
    }
}

// probe_k[m] = fft2(probe_real[m] + i*probe_imag[m])
__global__ void probe_fft_kernel(const float* __restrict__ pr,
                                 const float* __restrict__ pi_,
                                 float2* __restrict__ probeK)
{
    __shared__ float  tileRe[16][NPAD];
    __shared__ float  tileIm[16][NPAD];
    __shared__ float2 fFre[4][32];
    __shared__ float2 fFim[2][4][32];
    __shared__ float2 twT[2][8][32];
    fft_tables_init(fFre, fFim, twT);

    const int m = blockIdx.x;
    float2* img = probeK + (size_t)m * NPP;
    for (int p = threadIdx.x; p < NPP; p += blockDim.x) {
        float2 v;
        v.x = pr[(size_t)m * NPP + p];
        v.y = pi_[(size_t)m * NPP + p];
        img[p] = v;
    }
    __threadfence_block();
    __syncthreads();
    fft2_block(img, 0, 1.0f, tileRe, tileIm, fFre, fFim, twT);
}

// Each block processes one scan position b and mpb probe modes.
__global__ void ptycho_kernel(const float* __restrict__ obj_amp,
                              const float* __restrict__ obj_phase,
                              const float2* __restrict__ probeK,
                              const float2* __restrict__ propT,
                              const float* __restrict__ positions,
                              const int* __restrict__ indices,
                              float2* __restrict__ bufs,
                              float* __restrict__ out,
                              int mpb)
{
    __shared__ float  tileRe[16][NPAD];
    __shared__ float  tileIm[16][NPAD];
    __shared__ float2 fFre[4][32];
    __shared__ float2 fFim[2][4][32];
    __shared__ float2 twT[2][8][32];
    __shared__ float2 shx[NP];   // separable sub-pixel shift phases
    __shared__ float2 shy[NP];

    const int gpb = NMODES / mpb;
    const int b   = blockIdx.x / gpb;
    const int m0  = (blockIdx.x % gpb) * mpb;
    float2* img   = bufs + (size_t)blockIdx.x * NPP;
    float* outb   = out + (size_t)b * NPP;

    const int   idx = indices[b];
    const float p0  = positions[2 * idx + 0];
    const float p1  = positions[2 * idx + 1];
    const float ip0 = rintf(p0);
    const float ip1 = rintf(p1);
    const float d0  = (p0 - ip0) * PIXELSZ;
    const float d1  = (p1 - ip1) * PIXELSZ;
    const int   i0  = (int)ip0;
    const int   i1  = (int)ip1;

    fft_tables_init(fFre, fFim, twT);
    for (int i = threadIdx.x; i < NP; i += blockDim.x) {
        float s, c;
        __sincosf(-2.0f * PI_F * sfreq(i) * d0, &s, &c);
        shx[i] = make_float2(c, s);
        __sincosf(-2.0f * PI_F * sfreq(i) * d1, &s, &c);
        shy[i] = make_float2(c, s);
    }
    __syncthreads();

    for (int m = m0; m < m0 + mpb; ++m) {
        // wave_k = probe_k[m] * shift_phase; then ifft2
        for (int p = threadIdx.x; p < NPP; p += blockDim.x) {
            const int i = p >> 8, j = p & 255;
            float2 a = shx[i], bph = shy[j];
            float sr = a.x * bph.x - a.y * bph.y;
            float si = a.x * bph.y + a.y * bph.x;
            float2 q = probeK[(size_t)m * NPP + p];
            float2 w;
            w.x = q.x * sr - q.y * si;
            w.y = q.x * si + q.y * sr;
            img[p] = w;
        }
        __threadfence_block();
        __syncthreads();
        fft2_block(img, 1, 1.0f / 65536.0f, tileRe, tileIm, fFre, fFim, twT);

        // multiply by object slice 0 patch: amp * exp(i*phase)
        for (int p = threadIdx.x; p < NPP; p += blockDim.x) {
            const int i = p >> 8, j = p & 255;
            const size_t oidx = (size_t)(i0 + i) * NOBJ + (i1 + j);
            float a  = obj_amp[oidx];
            float ph = obj_phase[oidx];
            float s, c;
            __sincosf(ph, &s, &c);
            float orr = a * c, oii = a * s;
            float2 w = img[p];
            float2 r;
            r.x = w.x * orr - w.y * oii;
            r.y = w.x * oii + w.y * orr;
            img[p] = r;
        }
        __threadfence_block();
        __syncthreads();

        // multislice: wave = patch[s] * ifft2(fft2(wave) * prop)
        for (int sl = 1; sl < NSLICES; ++sl) {
            fft2_block(img, 0, 1.0f, tileRe, tileIm, fFre, fFim, twT);
            for (int p = threadIdx.x; p < NPP; p += blockDim.x) {
                float2 pr = propT[p];
                float2 w = img[p];
                float2 r;
                r.x = w.x * pr.x - w.y * pr.y;
                r.y = w.x * pr.y + w.y * pr.x;
                img[p] = r;
            }
            __threadfence_block();
            __syncthreads();
            fft2_block(img, 1, 1.0f / 65536.0f, tileRe, tileIm, fFre, fFim, twT);
            for (int p = threadIdx.x; p < NPP; p += blockDim.x) {
                const int i = p >> 8, j = p & 255;
                const size_t oidx = ((size_t)sl * NOBJ + (i0 + i)) * NOBJ + (i1 + j);
                float a  = obj_amp[oidx];
                float ph = obj_phase[oidx];
                float s, c;
                __sincosf(ph, &s, &c);
                float orr = a * c, oii = a * s;
                float2 w = img[p];
                float2 r;
                r.x = w.x * orr - w.y * oii;
                r.y = w.x * oii + w.y * orr;
                img[p] = r;
            }
            __threadfence_block();
            __syncthreads();
        }

        // diff = fftshift(fft2(wave)); accumulate |diff|^2 over modes
        fft2_block(img, 0, 1.0f, tileRe, tileIm, fFre, fFim, twT);
        for (int p = threadIdx.x; p < NPP; p += blockDim.x) {
            const int i = p >> 8, j = p & 255;
            const int oi = (i + NP / 2) & (NP - 1);
            const int oj = (j + NP / 2) & (NP - 1);
            float2 w = img[p];
            atomicAdd(&outb[oi * NP + oj], w.x * w.x + w.y * w.y);
        }
        __threadfence_block();
        __syncthreads();
    }
}

// ---------------------------------------------------------------------------
extern "C" void kernel_launch(void* const* d_in, const int* in_sizes, int n_in,
                              void* d_out, int out_size, void* d_ws, size_t ws_size,
                              hipStream_t stream)
{
    (void)in_sizes; (void)n_in; (void)out_size;

    const float* obj_amp    = (const float*)d_in[0];
    const float* obj_phase  = (const float*)d_in[1];
    const float* probe_real = (const float*)d_in[2];
    const float* probe_imag = (const float*)d_in[3];
    const float* positions  = (const float*)d_in[4];
    const int*   indices    = (const int*)d_in[5];
    float*       out        = (float*)d_out;

    const size_t imgBytes = (size_t)NPP * sizeof(float2);       // 512 KB
    float2* propT  = (float2*)d_ws;
    float2* probeK = (float2*)((char*)d_ws + imgBytes);
    float2* bufs   = (float2*)((char*)d_ws + (size_t)(1 + NMODES) * imgBytes);

    // (b,m)-parallel if scratch allows (256 blocks, ~131MB, L2-resident);
    // otherwise b-parallel with modes looped in-block (64 blocks, ~35MB).
    const size_t needBig = (size_t)(1 + NMODES + BATCH * NMODES) * imgBytes;
    const int mpb     = (ws_size >= needBig) ? 1 : NMODES;
    const int nblocks = BATCH * (NMODES / mpb);

    init_kernel<<<256, 256, 0, stream>>>(out, BATCH * NPP, propT);
    probe_fft_kernel<<<NMODES, 512, 0, stream>>>(probe_real, probe_imag, probeK);
    ptycho_kernel<<<nblocks, 512, 0, stream>>>(obj_amp, obj_phase, probeK, propT,
                                               positions, indices, bufs, out, mpb);
}